// VAERM_53618371723944
// MI455X (gfx1250) — compile-verified
//
#include <hip/hip_runtime.h>

typedef __attribute__((ext_vector_type(2))) float v2f;
typedef __attribute__((ext_vector_type(8))) float v8f;

#define BN 1024
#define DN 3
#define LN 3
#define HEN 256
#define HDN 128
#define NSTEPS 100

// ---- output offsets (flat concat of the 12-tuple) ----
#define OFF_Z        0
#define OFF_MUXZ     3072
#define OFF_LSXZ     6144
#define OFF_MUZX     9216
#define OFF_LTZX     12288
#define OFF_MUZ      15360
#define OFF_LTZ      18432
#define OFF_DW       19456
#define OFF_DT       22528
#define OFF_ZPRIOR   25600
#define OFF_DWPRIOR  28672
#define OFF_DTPRIOR  31744

// Fast sigmoid: v_exp_f32 + v_add + v_rcp_f32 (no IEEE-division fixup chain).
__device__ __forceinline__ float sigf(float x) {
  return __builtin_amdgcn_rcpf(1.0f + __expf(-x));
}

// Sum across the aligned 4-lane group via DPP quad_perm (pure VALU, no LDS pipe).
__device__ __forceinline__ float qsum4(float v) {
#if __has_builtin(__builtin_amdgcn_update_dpp)
  int t = __builtin_amdgcn_update_dpp(0, __float_as_int(v), 0xB1, 0xF, 0xF, true); // [1,0,3,2]
  v += __int_as_float(t);
  t = __builtin_amdgcn_update_dpp(0, __float_as_int(v), 0x4E, 0xF, 0xF, true);     // [2,3,0,1]
  v += __int_as_float(t);
#else
  v += __shfl_xor(v, 1, 32);
  v += __shfl_xor(v, 2, 32);
#endif
  return v;
}

// ---------------- encoder layer 1: h1 = swish(x @ W1 + b1), K=3 ----------------
__global__ void k_enc1(const float* __restrict__ x, const float* __restrict__ W1,
                       const float* __restrict__ b1, float* __restrict__ h1) {
  int i = blockIdx.x * blockDim.x + threadIdx.x;
  if (i >= BN * HEN) return;
  int b = i / HEN, e = i % HEN;
  float acc = b1[e];
#pragma unroll
  for (int d = 0; d < 3; ++d) acc = fmaf(x[b * 3 + d], W1[d * HEN + e], acc);
  h1[i] = acc * sigf(acc);
}

// ------- encoder layer 2: h2 = swish(h1 @ W2 + b2) via V_WMMA_F32_16X16X4_F32 -------
// One wave per 16x16 output tile; 64 k-steps of 4.
// A (16x4 f32): v0 = K0 (lanes 0-15) / K2 (lanes 16-31), v1 = K1 / K3.
// B (4x16 f32): mirrored; C/D: VGPR r -> (M=r, N=lane) | (M=r+8, N=lane-16).
__global__ void k_enc2_wmma(const float* __restrict__ h1, const float* __restrict__ W2,
                            const float* __restrict__ b2, float* __restrict__ h2) {
  int wave = (blockIdx.x * blockDim.x + threadIdx.x) >> 5;  // 0..1023
  int lane = threadIdx.x & 31;
  int tm = wave >> 4;            // 0..63 (M tiles)
  int tn = wave & 15;            // 0..15 (N tiles)
  int half = lane >> 4;          // 0 | 1
  int l15 = lane & 15;
  int arow = tm * 16 + l15;
  int bcol = tn * 16 + l15;

  v8f c = {};
  for (int k0 = 0; k0 < HEN; k0 += 4) {
    int ka = k0 + half * 2;
    v2f a, b;
    a.x = h1[arow * HEN + ka];
    a.y = h1[arow * HEN + ka + 1];
    b.x = W2[ka * HEN + bcol];
    b.y = W2[(ka + 1) * HEN + bcol];
    c = __builtin_amdgcn_wmma_f32_16x16x4_f32(false, a, false, b, (short)0, c,
                                              false, false);
  }
  float bias = b2[bcol];
#pragma unroll
  for (int r = 0; r < 8; ++r) {
    int row = tm * 16 + r + half * 8;
    float v = c[r] + bias;
    h2[row * HEN + bcol] = v * sigf(v);
  }
}

// ---------------- encoder heads: mu_zx, log_t_zx (K=256, N=3) ----------------
__global__ void k_enc_heads(const float* __restrict__ h2,
                            const float* __restrict__ muW, const float* __restrict__ mub,
                            const float* __restrict__ ltW, const float* __restrict__ ltb,
                            float* __restrict__ out) {
  int i = blockIdx.x * blockDim.x + threadIdx.x;
  if (i >= BN * LN) return;
  int b = i / LN, l = i % LN;
  float am = mub[l], at = ltb[l];
  const float* hr = h2 + b * HEN;
  for (int j = 0; j < HEN; ++j) {
    float h = hr[j];
    am = fmaf(h, muW[j * LN + l], am);
    at = fmaf(h, ltW[j * LN + l], at);
  }
  out[OFF_MUZX + i] = am;
  out[OFF_LTZX + i] = at;
}

// ---------------- SDE integrator: both chains, 4 lanes per sample ----------------
__global__ void k_sde(const float* __restrict__ eps, const float* __restrict__ eps_prior,
                      const float* __restrict__ decW1, const float* __restrict__ decb1,
                      const float* __restrict__ decMuW,
                      const float* __restrict__ prior_mu_b,
                      const float* __restrict__ prior_lt_b,
                      float* __restrict__ out) {
  __shared__ float W1s[3 * HDN];
  __shared__ float b1s[HDN];
  __shared__ float Wms[HDN * 3];
  for (int i = threadIdx.x; i < 3 * HDN; i += blockDim.x) W1s[i] = decW1[i];
  for (int i = threadIdx.x; i < HDN; i += blockDim.x) b1s[i] = decb1[i];
  for (int i = threadIdx.x; i < HDN * 3; i += blockDim.x) Wms[i] = decMuW[i];
  __syncthreads();

  int gid = blockIdx.x * blockDim.x + threadIdx.x;
  int sg = gid >> 2;        // sample-group 0..2047
  int lg = gid & 3;         // lane within group
  int chain = sg >> 10;     // 0 = posterior, 1 = prior
  int b = sg & (BN - 1);

  float z[3], dt[3], sdt[3];
  if (chain == 0) {
#pragma unroll
    for (int l = 0; l < 3; ++l) {
      z[l] = out[OFF_MUZX + b * 3 + l];
      float lt = out[OFF_LTZX + b * 3 + l];
      dt[l] = __expf(2.0f * lt) * (1.0f / NSTEPS);
      sdt[l] = sqrtf(dt[l]);
    }
  } else {
    float lt = prior_lt_b[0];
    float d = __expf(2.0f * lt) * (1.0f / NSTEPS);
    float sd = sqrtf(d);
#pragma unroll
    for (int l = 0; l < 3; ++l) { z[l] = prior_mu_b[l]; dt[l] = d; sdt[l] = sd; }
  }
  const float* ee = chain ? eps_prior : eps;
  float dw[3] = {0.0f, 0.0f, 0.0f};

  const int PP[3][3] = {{0, 1, 2}, {1, 3, 4}, {2, 4, 5}};  // sym pair index

  for (int n = 0; n < NSTEPS; ++n) {
#pragma unroll
    for (int l = 0; l < 3; ++l) dw[l] = sdt[l] * ee[(n * BN + b) * 3 + l];
    if (n + 1 < NSTEPS) {
      // cover next step's only global load (emits global_prefetch_b8)
      __builtin_prefetch(&ee[((n + 1) * BN + b) * 3], 0, 3);
    }

    float JF[3][3];
    float Hs[3][6];
#pragma unroll
    for (int d = 0; d < 3; ++d) {
#pragma unroll
      for (int l = 0; l < 3; ++l) JF[d][l] = 0.0f;
#pragma unroll
      for (int p = 0; p < 6; ++p) Hs[d][p] = 0.0f;
    }

    // 32 of 128 hidden units per lane
    for (int t = 0; t < 32; ++t) {
      int j = lg * 32 + t;
      float w0 = W1s[j], w1 = W1s[HDN + j], w2 = W1s[2 * HDN + j];
      float pre = fmaf(z[0], w0, fmaf(z[1], w1, fmaf(z[2], w2, b1s[j])));
      float sg_ = sigf(pre);
      float om = 1.0f - sg_;
      float sp = sg_ * (1.0f + pre * om);                        // swish'
      float spp = sg_ * om * (2.0f + pre * (1.0f - 2.0f * sg_)); // swish''
      float A0 = sp * w0, A1 = sp * w1, A2 = sp * w2;
      float c00 = spp * w0 * w0, c01 = spp * w0 * w1, c02 = spp * w0 * w2;
      float c11 = spp * w1 * w1, c12 = spp * w1 * w2, c22 = spp * w2 * w2;
#pragma unroll
      for (int d = 0; d < 3; ++d) {
        float m = Wms[j * 3 + d];
        JF[d][0] = fmaf(m, A0, JF[d][0]);
        JF[d][1] = fmaf(m, A1, JF[d][1]);
        JF[d][2] = fmaf(m, A2, JF[d][2]);
        Hs[d][0] = fmaf(m, c00, Hs[d][0]);
        Hs[d][1] = fmaf(m, c01, Hs[d][1]);
        Hs[d][2] = fmaf(m, c02, Hs[d][2]);
        Hs[d][3] = fmaf(m, c11, Hs[d][3]);
        Hs[d][4] = fmaf(m, c12, Hs[d][4]);
        Hs[d][5] = fmaf(m, c22, Hs[d][5]);
      }
    }

    // reduce the 27 partials across the 4 lanes of the group (DPP quad_perm)
#pragma unroll
    for (int d = 0; d < 3; ++d) {
#pragma unroll
      for (int l = 0; l < 3; ++l) JF[d][l] = qsum4(JF[d][l]);
#pragma unroll
      for (int p = 0; p < 6; ++p) Hs[d][p] = qsum4(Hs[d][p]);
    }

    // G = JF^T JF (symmetric)
    float G[3][3];
#pragma unroll
    for (int k = 0; k < 3; ++k)
#pragma unroll
      for (int l = 0; l < 3; ++l)
        G[k][l] = JF[0][k] * JF[0][l] + JF[1][k] * JF[1][l] + JF[2][k] * JF[2][l];

    float det = G[0][0] * (G[1][1] * G[2][2] - G[1][2] * G[2][1])
              - G[0][1] * (G[1][0] * G[2][2] - G[1][2] * G[2][0])
              + G[0][2] * (G[1][0] * G[2][1] - G[1][1] * G[2][0]);
    float id = __builtin_amdgcn_rcpf(det);
    float Gi[3][3];
    Gi[0][0] = (G[1][1] * G[2][2] - G[1][2] * G[2][1]) * id;
    Gi[0][1] = (G[0][2] * G[2][1] - G[0][1] * G[2][2]) * id;
    Gi[0][2] = (G[0][1] * G[1][2] - G[0][2] * G[1][1]) * id;
    Gi[1][0] = (G[1][2] * G[2][0] - G[1][0] * G[2][2]) * id;
    Gi[1][1] = (G[0][0] * G[2][2] - G[0][2] * G[2][0]) * id;
    Gi[1][2] = (G[0][2] * G[1][0] - G[0][0] * G[1][2]) * id;
    Gi[2][0] = (G[1][0] * G[2][1] - G[1][1] * G[2][0]) * id;
    Gi[2][1] = (G[0][1] * G[2][0] - G[0][0] * G[2][1]) * id;
    Gi[2][2] = (G[0][0] * G[1][1] - G[0][1] * G[1][0]) * id;

    // DG[k][l][m] = dG_kl/dz_m = sum_d (H[d][k][m]*JF[d][l] + JF[d][k]*H[d][l][m])
    float DG[3][3][3];
#pragma unroll
    for (int k = 0; k < 3; ++k)
#pragma unroll
      for (int l = 0; l < 3; ++l)
#pragma unroll
        for (int m = 0; m < 3; ++m) {
          float s = 0.0f;
#pragma unroll
          for (int d = 0; d < 3; ++d)
            s += Hs[d][PP[k][m]] * JF[d][l] + JF[d][k] * Hs[d][PP[l][m]];
          DG[k][l][m] = s;
        }

    // detr[i] = 0.25 * sum_m Gi[i][m]*(2*S1[m]-S2[m]),
    // S1[m] = sum_{j,k} Gi[j][k]*DG[j][m][k], S2[m] = sum_{j,k} Gi[j][k]*DG[j][k][m]
    float S1[3], S2[3];
#pragma unroll
    for (int m = 0; m < 3; ++m) {
      float s1 = 0.0f, s2 = 0.0f;
#pragma unroll
      for (int j = 0; j < 3; ++j)
#pragma unroll
        for (int k = 0; k < 3; ++k) {
          s1 += Gi[j][k] * DG[j][m][k];
          s2 += Gi[j][k] * DG[j][k][m];
        }
      S1[m] = s1; S2[m] = s2;
    }

#pragma unroll
    for (int i = 0; i < 3; ++i) {
      float detr = 0.25f * (Gi[i][0] * (2.0f * S1[0] - S2[0]) +
                            Gi[i][1] * (2.0f * S1[1] - S2[1]) +
                            Gi[i][2] * (2.0f * S1[2] - S2[2]));
      float stoch = Gi[i][0] * dw[0] + Gi[i][1] * dw[1] + Gi[i][2] * dw[2];
      z[i] += detr + stoch;
    }
  }

  if (lg == 0) {
    int zo = chain ? OFF_ZPRIOR : OFF_Z;
    int dwo = chain ? OFF_DWPRIOR : OFF_DW;
    int dto = chain ? OFF_DTPRIOR : OFF_DT;
#pragma unroll
    for (int l = 0; l < 3; ++l) {
      out[zo + b * 3 + l] = z[l];
      out[dwo + b * 3 + l] = dw[l];
      out[dto + b * 3 + l] = dt[l];
    }
    if (chain) {
#pragma unroll
      for (int l = 0; l < 3; ++l) out[OFF_MUZ + b * 3 + l] = prior_mu_b[l];
      out[OFF_LTZ + b] = prior_lt_b[0];
    }
  }
}

// ---------------- decoder heads on posterior z ----------------
__global__ void k_dec(const float* __restrict__ decW1, const float* __restrict__ decb1,
                      const float* __restrict__ muW, const float* __restrict__ mub,
                      const float* __restrict__ lsW, const float* __restrict__ lsb,
                      float* __restrict__ out) {
  int b = blockIdx.x * blockDim.x + threadIdx.x;
  if (b >= BN) return;
  float z0 = out[OFF_Z + b * 3 + 0];
  float z1 = out[OFF_Z + b * 3 + 1];
  float z2 = out[OFF_Z + b * 3 + 2];
  float mu[3] = {mub[0], mub[1], mub[2]};
  float ls[3] = {lsb[0], lsb[1], lsb[2]};
  for (int j = 0; j < HDN; ++j) {
    float pre = fmaf(z0, decW1[j], fmaf(z1, decW1[HDN + j], fmaf(z2, decW1[2 * HDN + j], decb1[j])));
    float h = pre * sigf(pre);
#pragma unroll
    for (int d = 0; d < 3; ++d) {
      mu[d] = fmaf(h, muW[j * 3 + d], mu[d]);
      ls[d] = fmaf(h, lsW[j * 3 + d], ls[d]);
    }
  }
#pragma unroll
  for (int d = 0; d < 3; ++d) {
    out[OFF_MUXZ + b * 3 + d] = mu[d];
    out[OFF_LSXZ + b * 3 + d] = ls[d];
  }
}

extern "C" void kernel_launch(void* const* d_in, const int* in_sizes, int n_in,
                              void* d_out, int out_size, void* d_ws, size_t ws_size,
                              hipStream_t stream) {
  const float* x          = (const float*)d_in[0];
  const float* eps        = (const float*)d_in[1];
  const float* eps_prior  = (const float*)d_in[2];
  const float* enc_W1     = (const float*)d_in[3];
  const float* enc_b1     = (const float*)d_in[4];
  const float* enc_W2     = (const float*)d_in[5];
  const float* enc_b2     = (const float*)d_in[6];
  const float* enc_mu_W   = (const float*)d_in[7];
  const float* enc_mu_b   = (const float*)d_in[8];
  const float* enc_lt_W   = (const float*)d_in[9];
  const float* enc_lt_b   = (const float*)d_in[10];
  const float* dec_W1     = (const float*)d_in[11];
  const float* dec_b1     = (const float*)d_in[12];
  const float* dec_mu_W   = (const float*)d_in[13];
  const float* dec_mu_b   = (const float*)d_in[14];
  const float* dec_ls_W   = (const float*)d_in[15];
  const float* dec_ls_b   = (const float*)d_in[16];
  const float* prior_mu_b = (const float*)d_in[17];
  const float* prior_lt_b = (const float*)d_in[18];

  float* out = (float*)d_out;
  float* h1 = (float*)d_ws;            // 1024*256 f32 = 1 MB
  float* h2 = h1 + BN * HEN;           // 1 MB

  k_enc1<<<(BN * HEN + 255) / 256, 256, 0, stream>>>(x, enc_W1, enc_b1, h1);
  k_enc2_wmma<<<128, 256, 0, stream>>>(h1, enc_W2, enc_b2, h2);   // 1024 waves = 1024 tiles
  k_enc_heads<<<(BN * LN + 255) / 256, 256, 0, stream>>>(h2, enc_mu_W, enc_mu_b,
                                                         enc_lt_W, enc_lt_b, out);
  k_sde<<<32, 256, 0, stream>>>(eps, eps_prior, dec_W1, dec_b1, dec_mu_W,
                                prior_mu_b, prior_lt_b, out);
  k_dec<<<(BN + 255) / 256, 256, 0, stream>>>(dec_W1, dec_b1, dec_mu_W, dec_mu_b,
                                              dec_ls_W, dec_ls_b, out);
}